// NeuralMem_60894046323322
// MI455X (gfx1250) — compile-verified
//
#include <hip/hip_runtime.h>

typedef __attribute__((ext_vector_type(2))) float v2f;
typedef __attribute__((ext_vector_type(8))) float v8f;

#define KDIM 75
#define KPAD 80      // pad K 75 -> 80 (20 WMMA k-steps of 4); zeros don't affect dot
#define LDIM 112     // output lattice per side
#define NQ   (LDIM*LDIM)   // 12544 query patches
#define NCB  16384
#define HH 96
#define WW 96
#define CCH 3
#define PADW 116     // 96 + 2*10

#define STAGE_ELEMS (16 * KPAD)   // 1280 floats per B tile
#define NITER (NCB / 16)          // 1024 N-chunks

#if defined(__AMDGCN__) && __has_builtin(__builtin_amdgcn_global_load_async_to_lds_b32) && \
    __has_builtin(__builtin_amdgcn_s_wait_asynccnt)
#define USE_ASYNC_LDS 1
#else
#define USE_ASYNC_LDS 0
#endif

#define GLOBAL_AS __attribute__((address_space(1)))
#define LDS_AS    __attribute__((address_space(3)))

// ---------------------------------------------------------------- build q
__global__ void nm_build_q(const float* __restrict__ img, float* __restrict__ qPad) {
    int t = blockIdx.x * blockDim.x + threadIdx.x;
    if (t >= NQ * KPAD) return;
    int p = t / KPAD, k = t % KPAD;
    float v = 0.0f;
    if (k < KDIM) {
        int c  = k / 25, rem = k % 25, ki = rem / 5, kj = rem % 5;
        int i0 = p / LDIM, i1 = p % LDIM;
        int h = i0 + ki - 10, w = i1 + kj - 10;
        if (h >= 0 && h < HH && w >= 0 && w < WW)
            v = img[(h * WW + w) * CCH + c];
    }
    qPad[t] = v;
}

// ----------------------------------------------- pad codebook + row norms
__global__ void nm_build_cb(const float* __restrict__ cb, float* __restrict__ cbPad,
                            float* __restrict__ cnorm) {
    int n = blockIdx.x * blockDim.x + threadIdx.x;
    if (n >= NCB) return;
    float s = 0.0f;
    for (int k = 0; k < KPAD; ++k) {
        float v = (k < KDIM) ? cb[n * KDIM + k] : 0.0f;
        cbPad[n * KPAD + k] = v;
        s += v * v;
    }
    cnorm[n] = s;
}

// ------------------------------------------------------- init accumulator
__global__ void nm_init(float* __restrict__ acc, unsigned* __restrict__ gmax) {
    int t = blockIdx.x * blockDim.x + threadIdx.x;
    if (t < CCH * PADW * PADW) acc[t] = 0.0f;
    if (t == 0) *gmax = 0u;   // float bits of +0.0; all data >= 0
}

// -------------------------------------- fused GEMM (WMMA f32) + argmin
// 256 threads = 8 waves; wave w owns M-tile (blockIdx.x*8 + w)*16.
// B tile (16 codebook rows x 80 f32) double-buffered in LDS via
// GLOBAL_LOAD_ASYNC_TO_LDS_B32 (ASYNCcnt), shared by all 8 waves.
__global__ __launch_bounds__(256)
void nm_gemm_argmin(const float* __restrict__ qPad, const float* __restrict__ cbPad,
                    const float* __restrict__ cnorm, int* __restrict__ idxOut) {
    __shared__ __align__(16) float ldsB[2][STAGE_ELEMS];   // 2 x 5 KB

    int tid  = threadIdx.x;
    int wave = tid >> 5, lane = tid & 31;
    int half = lane >> 4, l16 = lane & 15;
    int m0 = (blockIdx.x * 8 + wave) * 16;

    // A tile (16 x 80 f32) resident in registers: 20 x v2f per lane.
    // Lane l16 = row M, half selects k pair {2*half, 2*half+1} per 4-wide k-step.
    v2f a[20];
    const float* aRow = qPad + (size_t)(m0 + l16) * KPAD + half * 2;
#pragma unroll
    for (int kk = 0; kk < 20; ++kk)
        a[kk] = *(const v2f*)(aRow + kk * 4);

    float best[8];
    int   bidx[8];
#pragma unroll
    for (int v = 0; v < 8; ++v) { best[v] = 3.4e38f; bidx[v] = 0; }

#if USE_ASYNC_LDS
    // Each thread issues exactly 5 b32 async copies per stage -> uniform
    // per-wave ASYNCcnt; s_wait_asynccnt(0) before the barrier gates exactly
    // the current buffer (next stage not yet issued at that point).
    {
        const float* src = cbPad;                 // chunk 0
#pragma unroll
        for (int i = 0; i < 5; ++i) {
            int e = tid + i * 256;
            __builtin_amdgcn_global_load_async_to_lds_b32(
                (GLOBAL_AS int*)(unsigned long long)(src + e),
                (LDS_AS int*)(LDS_AS void*)(&ldsB[0][e]), 0, 0);
        }
    }
    for (int it = 0; it < NITER; ++it) {
        int cur = it & 1;
        int n0  = it * 16;
        __builtin_amdgcn_s_wait_asynccnt(0);      // my loads for buf[cur] done
        __syncthreads();                          // everyone's done; prev reads of buf[cur^1] done
        if (it + 1 < NITER) {                     // overlap next DMA with WMMA
            const float* src = cbPad + (size_t)(n0 + 16) * KPAD;
#pragma unroll
            for (int i = 0; i < 5; ++i) {
                int e = tid + i * 256;
                __builtin_amdgcn_global_load_async_to_lds_b32(
                    (GLOBAL_AS int*)(unsigned long long)(src + e),
                    (LDS_AS int*)(LDS_AS void*)(&ldsB[cur ^ 1][e]), 0, 0);
            }
        }
#else
    for (int it = 0; it < NITER; ++it) {
        int cur = 0;
        int n0  = it * 16;
        __syncthreads();                          // protect prior-iter LDS reads
        for (int e = tid; e < STAGE_ELEMS; e += 256)
            ldsB[0][e] = cbPad[(size_t)n0 * KPAD + e];
        __syncthreads();
#endif

        v8f c = {};
        const float* bRow = &ldsB[cur][0] + l16 * KPAD + half * 2; // col N = l16
#pragma unroll
        for (int kk = 0; kk < 20; ++kk) {
            v2f b = *(const v2f*)(bRow + kk * 4);
            c = __builtin_amdgcn_wmma_f32_16x16x4_f32(
                    /*neg_a=*/false, a[kk], /*neg_b=*/false, b,
                    /*c_mod=*/(short)0, c, /*reuse_a=*/false, /*reuse_b=*/false);
        }

        int   n  = n0 + l16;
        float cn = cnorm[n];
#pragma unroll
        for (int v = 0; v < 8; ++v) {
            float score = cn - 2.0f * c[v];       // ||c||^2 - 2 q.c
            if (score < best[v]) { best[v] = score; bidx[v] = n; }
        }
    }

    // Min-reduce across the 16 lanes holding one matrix row (width-16 xor tree).
#pragma unroll
    for (int v = 0; v < 8; ++v) {
        float bv = best[v];
        int   bi = bidx[v];
#pragma unroll
        for (int offs = 1; offs < 16; offs <<= 1) {
            float ov = __shfl_xor(bv, offs, 16);
            int   oi = __shfl_xor(bi, offs, 16);
            if (ov < bv || (ov == bv && oi < bi)) { bv = ov; bi = oi; }
        }
        if (l16 == 0) idxOut[m0 + v + half * 8] = bi;  // VGPR v: rows v / v+8
    }
}

// ----------------------------------------------- overlap-add reconstruction
__global__ void nm_scatter(const float* __restrict__ cb, const int* __restrict__ idx,
                           float* __restrict__ acc) {
    int t = blockIdx.x * blockDim.x + threadIdx.x;
    if (t >= NQ * KDIM) return;
    int p = t / KDIM, k = t % KDIM;
    int n = idx[p];
    float v = cb[(size_t)n * KDIM + k];
    int c  = k / 25, rem = k % 25, ki = rem / 5, kj = rem % 5;
    int i0 = p / LDIM, i1 = p % LDIM;
    atomicAdd(&acc[c * PADW * PADW + (i0 + ki) * PADW + (i1 + kj)], v);
}

// ------------------------------------------------------- global max (crop)
__global__ void nm_max(const float* __restrict__ acc, unsigned* __restrict__ gmax) {
    int t = blockIdx.x * blockDim.x + threadIdx.x;
    if (t >= CCH * HH * WW) return;
    int c = t / (HH * WW), r = t % (HH * WW), h = r / WW, w = r % WW;
    float v = acc[c * PADW * PADW + (h + 10) * PADW + (w + 10)];
    atomicMax(gmax, __float_as_uint(v));  // valid ordering for non-negative floats
}

// --------------------------------------------- normalize + HWC transpose
__global__ void nm_final(const float* __restrict__ acc, const unsigned* __restrict__ gmax,
                         float* __restrict__ out) {
    int t = blockIdx.x * blockDim.x + threadIdx.x;
    if (t >= HH * WW * CCH) return;
    int hw = t / CCH, c = t % CCH, h = hw / WW, w = hw % WW;
    float mx = __uint_as_float(*gmax);
    out[t] = acc[c * PADW * PADW + (h + 10) * PADW + (w + 10)] / mx;
}

extern "C" void kernel_launch(void* const* d_in, const int* in_sizes, int n_in,
                              void* d_out, int out_size, void* d_ws, size_t ws_size,
                              hipStream_t stream) {
    const float* img = (const float*)d_in[0];   // 96*96*3
    const float* cb  = (const float*)d_in[1];   // 16384*75
    float* out = (float*)d_out;                 // 96*96*3

    // workspace layout (~9.4 MB)
    float*    qPad  = (float*)d_ws;                 // NQ*KPAD
    float*    cbPad = qPad  + (size_t)NQ  * KPAD;   // NCB*KPAD
    float*    cnorm = cbPad + (size_t)NCB * KPAD;   // NCB
    int*      idxA  = (int*)(cnorm + NCB);          // NQ
    float*    acc   = (float*)(idxA + NQ);          // CCH*PADW*PADW
    unsigned* gmax  = (unsigned*)(acc + CCH * PADW * PADW);

    nm_build_q <<<(NQ * KPAD + 255) / 256, 256, 0, stream>>>(img, qPad);
    nm_build_cb<<<(NCB + 255) / 256,      256, 0, stream>>>(cb, cbPad, cnorm);
    nm_init    <<<(CCH * PADW * PADW + 255) / 256, 256, 0, stream>>>(acc, gmax);

    // 784 M-tiles of 16 rows; 8 waves (tiles) per workgroup -> 98 blocks
    nm_gemm_argmin<<<98, 256, 0, stream>>>(qPad, cbPad, cnorm, idxA);

    nm_scatter <<<(NQ * KDIM + 255) / 256,   256, 0, stream>>>(cb, idxA, acc);
    nm_max     <<<(CCH * HH * WW + 255) / 256, 256, 0, stream>>>(acc, gmax);
    nm_final   <<<(HH * WW * CCH + 255) / 256, 256, 0, stream>>>(acc, gmax, out);
}